// Decoder_43456479101023
// MI455X (gfx1250) — compile-verified
//
#include <hip/hip_runtime.h>
#include <math.h>

// ---------------------------------------------------------------------------
// GConvLSTM forward for MI455X (gfx1250, wave32, WMMA).
// Scatter-bound workload (~1.5-2GB gather/atomic traffic vs 4.5 GFLOP), so we
// use full-precision V_WMMA_F32_16X16X4_F32 and fuse everything around the
// aggregations. See analysis above.
// ---------------------------------------------------------------------------

typedef __attribute__((ext_vector_type(2))) float v2f;
typedef __attribute__((ext_vector_type(8))) float v8f;

#define FIN   8
#define HID   64
#define KTOT  72          // FIN + HID, also padded K for the head GEMM
#define KSTEPS 18         // KTOT / 4
#define GCOLS 256         // 4 gates * HID
#define NT_G  16          // GCOLS / 16
#define ROWS_PER_BLOCK 128
#define EPS 1e-5f

__device__ __forceinline__ float sigmoidf_(float x) { return 1.0f / (1.0f + expf(-x)); }

// ---------------- graph normalization ----------------
__global__ void k_init_deg(float* deg, int N) {
  int i = blockIdx.x * blockDim.x + threadIdx.x;
  if (i < N) deg[i] = 1.0f;  // self-loop weight
}

__global__ void k_acc_deg(float* deg, const int* dst, const float* w, int E) {
  int e = blockIdx.x * blockDim.x + threadIdx.x;
  if (e < E) atomicAdd(&deg[dst[e]], w[e]);
}

__global__ void k_dinv(float* deg, int N) {
  int i = blockIdx.x * blockDim.x + threadIdx.x;
  if (i < N) deg[i] = rsqrtf(deg[i]);  // deg >= 1 always
}

__global__ void k_norm(float* norm, const float* dinv, const int* src, const int* dst,
                       const float* w, int E) {
  int e = blockIdx.x * blockDim.x + threadIdx.x;
  if (e < E) norm[e] = dinv[dst[e]] * w[e] * dinv[src[e]];
}

// ---------------- AXH = agg([X | H0]) , row stride 72 ----------------
__global__ void k_init_axh(float* AXH, const float* X, const float* H,
                           const float* dinv, int N) {
  long t = (long)blockIdx.x * blockDim.x + threadIdx.x;
  long total = (long)N * KTOT;
  if (t >= total) return;
  int n = (int)(t / KTOT);
  int k = (int)(t - (long)n * KTOT);
  float d2 = dinv[n] * dinv[n];
  float v = (k < FIN) ? X[(long)n * FIN + k] : H[(long)n * HID + (k - FIN)];
  AXH[t] = d2 * v;
}

__global__ void k_scatter_axh(float* AXH, const float* X, const float* H,
                              const int* src, const int* dst, const float* norm, int E) {
  long t = (long)blockIdx.x * blockDim.x + threadIdx.x;
  long total = (long)E * KTOT;
  if (t >= total) return;
  int e = (int)(t / KTOT);
  int k = (int)(t - (long)e * KTOT);
  int s = src[e];
  float v = (k < FIN) ? X[(long)s * FIN + k] : H[(long)s * HID + (k - FIN)];
  atomicAdd(&AXH[(long)dst[e] * KTOT + k], norm[e] * v);
}

// ---------------- generic agg (init + edge scatter) ----------------
__global__ void k_init_agg(float* dstbuf, const float* srcf, const float* dinv,
                           int N, int F, int stride) {
  long t = (long)blockIdx.x * blockDim.x + threadIdx.x;
  long total = (long)N * stride;
  if (t >= total) return;
  int n = (int)(t / stride);
  int k = (int)(t - (long)n * stride);
  float d = dinv[n];
  dstbuf[t] = (k < F) ? d * d * srcf[t] : 0.0f;
}

__global__ void k_scatter_agg(float* dstbuf, const float* srcf, const int* src,
                              const int* dst, const float* norm, int E, int F, int stride) {
  long t = (long)blockIdx.x * blockDim.x + threadIdx.x;
  long total = (long)E * F;
  if (t >= total) return;
  int e = (int)(t / F);
  int k = (int)(t - (long)e * F);
  float v = norm[e] * srcf[(long)src[e] * stride + k];
  atomicAdd(&dstbuf[(long)dst[e] * stride + k], v);
}

// ---------------------------------------------------------------------------
// Fused gates kernel: G = AXH @ Bcat + bias  (one [N,72]x[72,256] WMMA GEMM,
// B = [Wx;Wh] for gates i,f,c,o stacked along columns), then the full LSTM
// pointwise update, 3 layernorms (shfl reductions), relu and skip concat.
// Block = 256 threads = 8 waves; each wave owns a 16-node M-tile and all 16
// N-tiles (128 fp32 accumulator VGPRs). B staged in dynamic LDS as per-lane
// pre-swizzled fragments: frag element ((s*16+t)*32 + lane)*2 + j holds
// B[4s + 2*(lane/16) + j][16t + lane%16].
// ---------------------------------------------------------------------------
__global__ __launch_bounds__(256)
void k_gates(const float* __restrict__ AXH, const float* __restrict__ C,
             const float* __restrict__ Wx, const float* __restrict__ Wh,
             const float* __restrict__ bx, const float* __restrict__ bh,
             const float* __restrict__ bg, const float* __restrict__ wc,
             const float* __restrict__ ln_h_s, const float* __restrict__ ln_h_b,
             const float* __restrict__ ln_c_s, const float* __restrict__ ln_c_b,
             const float* __restrict__ ln_o_s, const float* __restrict__ ln_o_b,
             const float* __restrict__ skip,
             float* __restrict__ hiddenOut, float* __restrict__ cellOut,
             float* __restrict__ Y, int N) {
  extern __shared__ float smem[];
  float* Bf = smem;             // 18 * 16 * 32 * 2 = 18432 floats
  float* bias = smem + 18432;   // 256 floats: bx + bh + bg

  int tid = threadIdx.x;
  for (int i = tid; i < KSTEPS * NT_G * 64; i += 256) {
    int s = i >> 10;            // / 1024
    int rem = i & 1023;
    int t = rem >> 6;
    int rem2 = rem & 63;
    int L = rem2 >> 1;
    int j = rem2 & 1;
    int k = 4 * s + 2 * (L >> 4) + j;   // 0..71
    int c = 16 * t + (L & 15);          // 0..255
    int g = c >> 6, h = c & 63;
    Bf[i] = (k < FIN) ? Wx[(g * FIN + k) * HID + h]
                      : Wh[(g * HID + (k - FIN)) * HID + h];
  }
  if (tid < GCOLS) bias[tid] = bx[tid] + bh[tid] + bg[tid];
  __syncthreads();

  int wave = tid >> 5, lane = tid & 31;
  int half = lane >> 4, lcol = lane & 15;
  int rowBase = blockIdx.x * ROWS_PER_BLOCK + wave * 16;

  int arow = rowBase + lcol;            // A-frag row = lane%16
  if (arow > N - 1) arow = N - 1;       // clamp (results discarded for OOB rows)
  const float* aptr = AXH + (long)arow * KTOT;

  v8f acc[NT_G];
#pragma unroll
  for (int t = 0; t < NT_G; ++t)
#pragma unroll
    for (int r = 0; r < 8; ++r) acc[t][r] = 0.0f;

  for (int s = 0; s < KSTEPS; ++s) {
    int k0 = 4 * s + 2 * half;
    v2f a = *(const v2f*)(aptr + k0);
    const float* bbase = Bf + s * 1024 + lane * 2;
#pragma unroll
    for (int t = 0; t < NT_G; ++t) {
      v2f b = *(const v2f*)(bbase + t * 64);
      acc[t] = __builtin_amdgcn_wmma_f32_16x16x4_f32(
          false, a, false, b, (short)0, acc[t], false, false);
    }
  }

  // pointwise LSTM + layernorms. D[m][c]: m = r + 8*half, c = 16*t + lcol.
#pragma unroll
  for (int r = 0; r < 8; ++r) {
    int node = rowBase + r + 8 * half;
    bool valid = node < N;
    int nc = valid ? node : N - 1;
    float hloc[4], cloc[4];
#pragma unroll
    for (int t = 0; t < 4; ++t) {
      int h = t * 16 + lcol;
      float c0 = C[(long)nc * HID + h];
      float gi = acc[t][r]      + bias[h]       + wc[h] * c0;
      float gf = acc[t + 4][r]  + bias[64 + h]  + wc[64 + h] * c0;
      float gc = acc[t + 8][r]  + bias[128 + h];
      float go = acc[t + 12][r] + bias[192 + h];
      float iv = sigmoidf_(gi);
      float fv = sigmoidf_(gf);
      float cc = fv * c0 + iv * tanhf(gc);
      float ov = sigmoidf_(go + wc[128 + h] * cc);
      hloc[t] = ov * tanhf(cc);
      cloc[t] = cc;
    }
    // layernorm stats over the node's 64 values: 4 accumulators x 16 lanes
    float sh = 0.f, sh2 = 0.f, sc = 0.f, sc2 = 0.f;
#pragma unroll
    for (int t = 0; t < 4; ++t) {
      sh += hloc[t]; sh2 += hloc[t] * hloc[t];
      sc += cloc[t]; sc2 += cloc[t] * cloc[t];
    }
#pragma unroll
    for (int mask = 1; mask < 16; mask <<= 1) {
      sh  += __shfl_xor(sh,  mask, 32);
      sh2 += __shfl_xor(sh2, mask, 32);
      sc  += __shfl_xor(sc,  mask, 32);
      sc2 += __shfl_xor(sc2, mask, 32);
    }
    float mh = sh * (1.0f / 64.0f), mc = sc * (1.0f / 64.0f);
    float vh = sh2 * (1.0f / 64.0f) - mh * mh;
    float vc = sc2 * (1.0f / 64.0f) - mc * mc;
    float rh = rsqrtf(vh + EPS), rc = rsqrtf(vc + EPS);
    if (valid) {
#pragma unroll
      for (int t = 0; t < 4; ++t) {
        int h = t * 16 + lcol;
        float hn = (hloc[t] - mh) * rh;
        float cn = (cloc[t] - mc) * rc;
        hiddenOut[(long)node * HID + h] = hn * ln_h_s[h] + ln_h_b[h];
        cellOut[(long)node * HID + h]   = cn * ln_c_s[h] + ln_c_b[h];
        float ov = hn * ln_o_s[h] + ln_o_b[h];
        Y[(long)node * KTOT + h] = ov > 0.0f ? ov : 0.0f;
      }
      if (lcol < 8) {  // cols 64,65 = skip, 66..71 = zero pad (K padding)
        Y[(long)node * KTOT + 64 + lcol] =
            (lcol < 2) ? skip[(long)node * 2 + lcol] : 0.0f;
      }
    }
  }
}

// ---------------------------------------------------------------------------
// Head GEMM: Z = relu(AY @ W1 + b1).  AY is [N,72] (cols 66..71 zero),
// W1 is [66,64] zero-padded to K=72.  4 N-tiles per wave.
// ---------------------------------------------------------------------------
__global__ __launch_bounds__(256)
void k_gemm2(const float* __restrict__ AY, const float* __restrict__ W1,
             const float* __restrict__ b1, float* __restrict__ Z, int N) {
  __shared__ float Bf[KSTEPS * 4 * 64];  // 4608 floats
  __shared__ float b1s[HID];

  int tid = threadIdx.x;
  for (int i = tid; i < KSTEPS * 4 * 64; i += 256) {
    int s = i >> 8;             // / 256
    int rem = i & 255;
    int t = rem >> 6;
    int rem2 = rem & 63;
    int L = rem2 >> 1;
    int j = rem2 & 1;
    int k = 4 * s + 2 * (L >> 4) + j;
    int c = 16 * t + (L & 15);
    Bf[i] = (k < HID + 2) ? W1[k * HID + c] : 0.0f;
  }
  if (tid < HID) b1s[tid] = b1[tid];
  __syncthreads();

  int wave = tid >> 5, lane = tid & 31;
  int half = lane >> 4, lcol = lane & 15;
  int rowBase = blockIdx.x * ROWS_PER_BLOCK + wave * 16;

  int arow = rowBase + lcol;
  if (arow > N - 1) arow = N - 1;
  const float* aptr = AY + (long)arow * KTOT;

  v8f acc[4];
#pragma unroll
  for (int t = 0; t < 4; ++t)
#pragma unroll
    for (int r = 0; r < 8; ++r) acc[t][r] = 0.0f;

  for (int s = 0; s < KSTEPS; ++s) {
    int k0 = 4 * s + 2 * half;
    v2f a = *(const v2f*)(aptr + k0);
    const float* bbase = Bf + s * 256 + lane * 2;
#pragma unroll
    for (int t = 0; t < 4; ++t) {
      v2f b = *(const v2f*)(bbase + t * 64);
      acc[t] = __builtin_amdgcn_wmma_f32_16x16x4_f32(
          false, a, false, b, (short)0, acc[t], false, false);
    }
  }

#pragma unroll
  for (int r = 0; r < 8; ++r) {
    int node = rowBase + r + 8 * half;
    if (node >= N) continue;
#pragma unroll
    for (int t = 0; t < 4; ++t) {
      int h = t * 16 + lcol;
      float z = acc[t][r] + b1s[h];
      Z[(long)node * HID + h] = z > 0.0f ? z : 0.0f;
    }
  }
}

// ---------------- final: out = AZ @ W2 + b2 + X[:,0] ----------------
__global__ void k_final(const float* __restrict__ AZ, const float* __restrict__ W2,
                        const float* __restrict__ b2, const float* __restrict__ X,
                        float* __restrict__ out, int N) {
  __shared__ float w2s[HID];
  if (threadIdx.x < HID) w2s[threadIdx.x] = W2[threadIdx.x];
  __syncthreads();
  int n = blockIdx.x * blockDim.x + threadIdx.x;
  if (n >= N) return;
  const float* a = AZ + (long)n * HID;
  float acc = b2[0] + X[(long)n * FIN];
#pragma unroll 8
  for (int h = 0; h < HID; ++h) acc += a[h] * w2s[h];
  out[n] = acc;
}

// ---------------------------------------------------------------------------
extern "C" void kernel_launch(void* const* d_in, const int* in_sizes, int n_in,
                              void* d_out, int out_size, void* d_ws, size_t ws_size,
                              hipStream_t stream) {
  const float* X    = (const float*)d_in[0];
  const int*   eidx = (const int*)d_in[1];
  const float* ew   = (const float*)d_in[2];
  const float* skip = (const float*)d_in[3];
  const float* H    = (const float*)d_in[4];
  const float* C    = (const float*)d_in[5];
  const float* Wx   = (const float*)d_in[6];
  const float* bx   = (const float*)d_in[7];
  const float* Wh   = (const float*)d_in[8];
  const float* bh   = (const float*)d_in[9];
  const float* wc   = (const float*)d_in[10];
  const float* bg   = (const float*)d_in[11];
  const float* ln_h_s = (const float*)d_in[12];
  const float* ln_h_b = (const float*)d_in[13];
  const float* ln_c_s = (const float*)d_in[14];
  const float* ln_c_b = (const float*)d_in[15];
  const float* ln_o_s = (const float*)d_in[16];
  const float* ln_o_b = (const float*)d_in[17];
  const float* W1   = (const float*)d_in[18];
  const float* b1   = (const float*)d_in[19];
  const float* W2   = (const float*)d_in[20];
  const float* b2   = (const float*)d_in[21];

  const int N = in_sizes[0] / FIN;
  const int E = in_sizes[2];

  const int* src = eidx;       // edge_index[0]
  const int* dst = eidx + E;   // edge_index[1]

  // outputs: out [N], hidden [N*64], cell [N*64]
  float* out_p    = (float*)d_out;
  float* hidden_p = out_p + N;
  float* cell_p   = hidden_p + (long)N * HID;

  // workspace layout (floats)
  float* ws   = (float*)d_ws;
  float* dinv = ws;                       // N
  float* norm = dinv + N;                 // E
  float* AXH  = norm + E;                 // N*72
  float* Y    = AXH;                      // alias: k_gates consumes AXH rows
                                          // before writing the same rows of Y
  float* AY   = AXH + (long)N * KTOT;     // N*72
  float* Z    = AY + (long)N * KTOT;      // N*64
  float* AZ   = Z + (long)N * HID;        // N*64

  const int B = 256;
  auto cdiv = [](long a, long b) { return (unsigned)((a + b - 1) / b); };

  // 1) symmetric GCN normalization
  k_init_deg<<<cdiv(N, B), B, 0, stream>>>(dinv, N);
  k_acc_deg<<<cdiv(E, B), B, 0, stream>>>(dinv, dst, ew, E);
  k_dinv<<<cdiv(N, B), B, 0, stream>>>(dinv, N);
  k_norm<<<cdiv(E, B), B, 0, stream>>>(norm, dinv, src, dst, ew, E);

  // 2) AXH = agg([X | H0])
  k_init_axh<<<cdiv((long)N * KTOT, B), B, 0, stream>>>(AXH, X, H, dinv, N);
  k_scatter_axh<<<cdiv((long)E * KTOT, B), B, 0, stream>>>(AXH, X, H, src, dst, norm, E);

  // 3) fused gates (WMMA) + LSTM + layernorms -> hidden, cell, Y
  size_t smem_gates = (size_t)(KSTEPS * NT_G * 64 + GCOLS) * sizeof(float); // 74752 B
  k_gates<<<cdiv(N, ROWS_PER_BLOCK), B, smem_gates, stream>>>(
      AXH, C, Wx, Wh, bx, bh, bg, wc,
      ln_h_s, ln_h_b, ln_c_s, ln_c_b, ln_o_s, ln_o_b,
      skip, hidden_p, cell_p, Y, N);

  // 4) AY = agg(Y) over 66 real features (stride 72, pad zeroed)
  k_init_agg<<<cdiv((long)N * KTOT, B), B, 0, stream>>>(AY, Y, dinv, N, HID + 2, KTOT);
  k_scatter_agg<<<cdiv((long)E * (HID + 2), B), B, 0, stream>>>(AY, Y, src, dst, norm,
                                                               E, HID + 2, KTOT);

  // 5) Z = relu(AY @ W1 + b1)  (WMMA)
  k_gemm2<<<cdiv(N, ROWS_PER_BLOCK), B, 0, stream>>>(AY, W1, b1, Z, N);

  // 6) AZ = agg(Z)
  k_init_agg<<<cdiv((long)N * HID, B), B, 0, stream>>>(AZ, Z, dinv, N, HID, HID);
  k_scatter_agg<<<cdiv((long)E * HID, B), B, 0, stream>>>(AZ, Z, src, dst, norm,
                                                          E, HID, HID);

  // 7) out = AZ @ W2 + b2 + X[:,0]
  k_final<<<cdiv(N, B), B, 0, stream>>>(AZ, W2, b2, X, out_p, N);
}